// RelGATLayer_28484223107262
// MI455X (gfx1250) — compile-verified
//
#include <hip/hip_runtime.h>
#include <cstdint>
#include <cstddef>

#define N_NODES  50000
#define N_EDGES  500000
#define DIM_IN   1152
#define DIM_OUT  200
#define HEADS    4
#define NREL     45
#define HO       800          // HEADS * DIM_OUT
#define EPSF     1e-16f
#define LDS_PAD  17           // 16 rows + 1 pad -> conflict-free column reads

typedef __attribute__((ext_vector_type(2))) float v2f;
typedef __attribute__((ext_vector_type(8))) float v8f;

// ---------------------------------------------------------------------------
// Init kernels
// ---------------------------------------------------------------------------
__global__ void init_misc_kernel(float* __restrict__ maxb, float* __restrict__ denb,
                                 float* __restrict__ biasb) {
    int i = blockIdx.x * blockDim.x + threadIdx.x;
    if (i < N_NODES * HEADS) {
        maxb[i] = -INFINITY;
        denb[i] = 0.0f;
    }
    if (i < N_NODES) biasb[i] = 0.0f;
}

__global__ void zero_out_kernel(float4* __restrict__ o) {
    const size_t total = (size_t)N_NODES * HO / 4;
    const size_t stride = (size_t)gridDim.x * blockDim.x;
    for (size_t i = (size_t)blockIdx.x * blockDim.x + threadIdx.x; i < total; i += stride)
        o[i] = make_float4(0.f, 0.f, 0.f, 0.f);
}

// ---------------------------------------------------------------------------
// Projection GEMM: proj[n, h*200+o] = sum_i emb[n,i] * W[h*200+o, i]
// One block = one 16-row strip of nodes. A strip staged in LDS (transposed,
// padded). 8 waves sweep the 50 column tiles; K-loop in steps of 4 through
// v_wmma_f32_16x16x4_f32 (full fp32 matrix path, wave32).
// ---------------------------------------------------------------------------
__global__ void __launch_bounds__(256)
proj_wmma_kernel(const float* __restrict__ emb, const float* __restrict__ W,
                 float* __restrict__ proj) {
    extern __shared__ float ldsA[];             // [DIM_IN][LDS_PAD] : ldsA[i*17 + row]
    const int tid = threadIdx.x;
    const int n0  = blockIdx.x << 4;            // 16 rows per block, 50000/16 = 3125 exact

    // Stage A tile (coalesced global reads, padded transposed LDS writes)
    for (int idx = tid; idx < 16 * DIM_IN; idx += 256) {
        int r = idx / DIM_IN;
        int i = idx - r * DIM_IN;
        ldsA[i * LDS_PAD + r] = emb[(size_t)(n0 + r) * DIM_IN + i];
    }
    __syncthreads();

    const int lane  = tid & 31;
    const int wv    = tid >> 5;                 // 8 waves per block
    const int laneN = lane & 15;                // row (A) / col (B,C) within tile
    const int koff  = (lane >> 4) << 1;         // lanes 0-15 hold K{0,1}, 16-31 hold K{2,3}
    const int rbase = (lane >> 4) << 3;         // C: VGPR v -> row v (lo lanes) / v+8 (hi lanes)

    for (int ct = wv; ct < HO / 16; ct += 8) {  // 50 col tiles round-robined over 8 waves
        const int c0 = ct << 4;
        const float* wrow = W + (size_t)(c0 + laneN) * DIM_IN;
        v8f acc = {0.f, 0.f, 0.f, 0.f, 0.f, 0.f, 0.f, 0.f};

        #pragma unroll 4
        for (int k0 = 0; k0 < DIM_IN; k0 += 4) {
            v2f a, b;
            a.x = ldsA[(k0 + koff    ) * LDS_PAD + laneN];
            a.y = ldsA[(k0 + koff + 1) * LDS_PAD + laneN];
            const float2 bb = *(const float2*)(wrow + k0 + koff);   // 8B aligned
            b.x = bb.x; b.y = bb.y;
            acc = __builtin_amdgcn_wmma_f32_16x16x4_f32(
                false, a, false, b, (short)0, acc, false, false);
        }

        #pragma unroll
        for (int v = 0; v < 8; ++v)
            proj[(size_t)(n0 + rbase + v) * HO + c0 + laneN] = acc[v];
    }
}

// ---------------------------------------------------------------------------
// Attention logits + segment max. One wave32 per edge.
// ---------------------------------------------------------------------------
__global__ void __launch_bounds__(256)
edge_logits_kernel(const float* __restrict__ proj, const int* __restrict__ ei,
                   const int* __restrict__ etype, const float* __restrict__ attn,
                   float* __restrict__ ebuf, float* __restrict__ maxb) {
    const int lane = threadIdx.x & 31;
    const int e = (blockIdx.x * blockDim.x + threadIdx.x) >> 5;
    if (e >= N_EDGES) return;

    const int src = ei[e];
    const int dst = ei[N_EDGES + e];
    const int t   = etype[e];
    const float4* ps = (const float4*)(proj + (size_t)src * HO);

    #pragma unroll
    for (int h = 0; h < HEADS; ++h) {
        const float4* av = (const float4*)(attn + ((size_t)h * NREL + t) * DIM_OUT);
        const float4* ph = ps + h * (DIM_OUT / 4);
        float s = 0.f;
        for (int j = lane; j < DIM_OUT / 4; j += 32) {      // 50 float4 chunks
            float4 p = ph[j];
            float4 a = av[j];
            s += p.x * a.x + p.y * a.y + p.z * a.z + p.w * a.w;
        }
        #pragma unroll
        for (int off = 16; off; off >>= 1) s += __shfl_xor(s, off, 32);
        if (lane == 0) {
            float v = s > 0.f ? s : 0.2f * s;               // leaky_relu(0.2)
            ebuf[(size_t)e * HEADS + h] = v;
            atomicMax(&maxb[(size_t)dst * HEADS + h], v);   // global_atomic_max_num_f32
        }
    }
}

// ---------------------------------------------------------------------------
// exp(e - max), denominator accumulation, relation-bias scatter.
// One thread per (edge, head).
// ---------------------------------------------------------------------------
__global__ void __launch_bounds__(256)
softmax_norm_kernel(float* __restrict__ ebuf, const int* __restrict__ ei,
                    const int* __restrict__ etype, const float* __restrict__ maxb,
                    float* __restrict__ denb, const float* __restrict__ rel_bias,
                    float* __restrict__ biasb) {
    int idx = blockIdx.x * blockDim.x + threadIdx.x;
    if (idx >= N_EDGES * HEADS) return;
    const int e = idx >> 2;         // HEADS == 4
    const int h = idx & 3;
    const int dst = ei[N_EDGES + e];
    float w = __expf(ebuf[idx] - maxb[dst * HEADS + h]);
    ebuf[idx] = w;
    atomicAdd(&denb[dst * HEADS + h], w);
    if (h == 0) atomicAdd(&biasb[dst], rel_bias[etype[e]]);
}

// ---------------------------------------------------------------------------
// Weighted message scatter: out[dst] += alpha * proj[src]. One wave per edge,
// lanes stride consecutive features -> coalesced f32 atomics (L2-resident).
// ---------------------------------------------------------------------------
__global__ void __launch_bounds__(256)
edge_scatter_kernel(const float* __restrict__ proj, const int* __restrict__ ei,
                    const float* __restrict__ ebuf, const float* __restrict__ denb,
                    float* __restrict__ out) {
    const int lane = threadIdx.x & 31;
    const int e = (blockIdx.x * blockDim.x + threadIdx.x) >> 5;
    if (e >= N_EDGES) return;

    const int src = ei[e];
    const int dst = ei[N_EDGES + e];

    float alpha[HEADS];
    #pragma unroll
    for (int h = 0; h < HEADS; ++h)
        alpha[h] = ebuf[(size_t)e * HEADS + h] /
                   fmaxf(denb[(size_t)dst * HEADS + h], EPSF);

    const float* ps = proj + (size_t)src * HO;
    float* po       = out  + (size_t)dst * HO;
    #pragma unroll
    for (int h = 0; h < HEADS; ++h) {
        const int base = h * DIM_OUT;
        for (int j = lane; j < DIM_OUT; j += 32)
            atomicAdd(&po[base + j], ps[base + j] * alpha[h]);
    }
}

// ---------------------------------------------------------------------------
// Broadcast per-node scalar bias onto all 800 features.
// ---------------------------------------------------------------------------
__global__ void __launch_bounds__(256)
bias_add_kernel(float* __restrict__ out, const float* __restrict__ biasb) {
    const float b = biasb[blockIdx.x];
    float* po = out + (size_t)blockIdx.x * HO;
    for (int j = threadIdx.x; j < HO; j += blockDim.x) po[j] += b;
}

// ---------------------------------------------------------------------------
// Launch
// ---------------------------------------------------------------------------
extern "C" void kernel_launch(void* const* d_in, const int* in_sizes, int n_in,
                              void* d_out, int out_size, void* d_ws, size_t ws_size,
                              hipStream_t stream) {
    const float* node_emb  = (const float*)d_in[0];   // [N, IN]
    const int*   edge_idx  = (const int*)  d_in[1];   // [2, E]
    const int*   edge_type = (const int*)  d_in[2];   // [E]
    const float* proj_w    = (const float*)d_in[3];   // [H, OUT, IN] = [800, 1152]
    const float* attn_vec  = (const float*)d_in[4];   // [H, R, OUT]
    const float* rel_bias  = (const float*)d_in[5];   // [R]
    float*       out       = (float*)d_out;           // [N, 800]

    // Workspace layout (all offsets 256B-aligned by construction)
    char* ws = (char*)d_ws;
    float* proj  = (float*)(ws);                        // 160,000,000 B
    float* ebuf  = (float*)(ws + 160000000);            //   8,000,000 B
    float* maxb  = (float*)(ws + 168000000);            //     800,000 B
    float* denb  = (float*)(ws + 168800000);            //     800,000 B
    float* biasb = (float*)(ws + 169600000);            //     200,000 B

    init_misc_kernel<<<(N_NODES * HEADS + 255) / 256, 256, 0, stream>>>(maxb, denb, biasb);
    zero_out_kernel<<<4096, 256, 0, stream>>>((float4*)out);

    // GEMM: 3125 blocks x 8 waves, 78,336 B dynamic LDS per block
    proj_wmma_kernel<<<N_NODES / 16, 256, DIM_IN * LDS_PAD * 4, stream>>>(
        node_emb, proj_w, proj);

    edge_logits_kernel<<<N_EDGES / 8, 256, 0, stream>>>(
        proj, edge_idx, edge_type, attn_vec, ebuf, maxb);

    softmax_norm_kernel<<<(N_EDGES * HEADS + 255) / 256, 256, 0, stream>>>(
        ebuf, edge_idx, edge_type, maxb, denb, rel_bias, biasb);

    edge_scatter_kernel<<<N_EDGES / 8, 256, 0, stream>>>(
        proj, edge_idx, ebuf, denb, out);

    bias_add_kernel<<<N_NODES, 256, 0, stream>>>(out, biasb);
}